// BaseGCN_62697932587513
// MI455X (gfx1250) — compile-verified
//
#include <hip/hip_runtime.h>
#include <math.h>

// ---------------------------------------------------------------------------
// GCN (3 layers) for MI455X / gfx1250.
//   h   = x @ W[l]^T + b[l]              -> fp32 WMMA (v_wmma_f32_16x16x4_f32)
//   agg = scatter_add(norm * h[row], col) -> wave-per-edge, f32 atomics (L2 resident)
//   x   = x + relu(BN(agg))               -> fused elementwise
// Whole working set (~84 MB) fits in the 192 MB L2 -> L2-bandwidth bound.
// ---------------------------------------------------------------------------

#define N_NODES 50000
#define N_EDGES 800000
#define DFEAT   128
#define N_LAYERS 3
#define BN_EPS  1e-5f

typedef __attribute__((ext_vector_type(2))) float v2f;
typedef __attribute__((ext_vector_type(8))) float v8f;

// -------------------------------------------------------------------- utils
__global__ void zero_f32(float* __restrict__ p, int n) {
    int i = blockIdx.x * blockDim.x + threadIdx.x;
    int stride = gridDim.x * blockDim.x;
    for (; i < n; i += stride) p[i] = 0.0f;
}

__global__ void degree_kernel(const int* __restrict__ row, float* __restrict__ deg, int E) {
    int i = blockIdx.x * blockDim.x + threadIdx.x;
    if (i < E) atomicAdd(&deg[row[i]], 1.0f);
}

__global__ void dinv_kernel(float* __restrict__ d, int N) {
    int i = blockIdx.x * blockDim.x + threadIdx.x;
    if (i < N) {
        float v = d[i];
        d[i] = (v > 0.0f) ? rsqrtf(v) : 0.0f;
    }
}

// ------------------------------------------------------------- WMMA GEMM
// H[M x 128] = X[M x 128] @ W^T + bias,  W is [128 x 128] row-major (W[n][k]).
// One wave computes one 16x16 tile of H with V_WMMA_F32_16X16X4_F32,
// looping K in 32 steps of 4.
//
// Per ISA 16x4 f32 A layout: lane L, vgpr v -> A[m0 + L%16, k0 + 2*(L/16) + v]
// B (4x16) mirrored:         lane L, vgpr v -> B[k0 + 2*(L/16) + v, n0 + L%16]
//   and B[k][n] = W[n][k], so both A and B are contiguous float2 loads.
// C layout: vgpr r -> H[m0 + r + 8*(L/16), n0 + L%16]; bias is per-column,
// i.e. constant per lane, so it directly initializes the accumulator.
__global__ void gemm_wmma(const float* __restrict__ X,
                          const float* __restrict__ W,
                          const float* __restrict__ bias,
                          float* __restrict__ H, int M) {
    const int lane = threadIdx.x & 31;
    const int wave = threadIdx.x >> 5;
    const int wavesPerBlock = blockDim.x >> 5;
    const int tilesN = DFEAT / 16;                 // 8
    const int tilesM = M >> 4;
    int tileId = blockIdx.x * wavesPerBlock + wave;
    if (tileId >= tilesM * tilesN) return;         // wave-uniform -> EXEC stays all-1s
    const int tm = tileId / tilesN;
    const int tn = tileId - tm * tilesN;
    const int m0 = tm << 4;
    const int n0 = tn << 4;

    const int l16   = lane & 15;
    const int khalf = (lane >> 4) << 1;            // 0 or 2

    const float* __restrict__ arow = X + (size_t)(m0 + l16) * DFEAT;
    const float* __restrict__ brow = W + (size_t)(n0 + l16) * DFEAT;

    const float bv = bias[n0 + l16];               // per-column bias, constant per lane
    v8f c = {bv, bv, bv, bv, bv, bv, bv, bv};

#pragma unroll
    for (int k0 = 0; k0 < DFEAT; k0 += 4) {
        v2f a = *(const v2f*)(arow + k0 + khalf);
        v2f b = *(const v2f*)(brow + k0 + khalf);
        // 8 args: (neg_a, A, neg_b, B, c_mod, C, reuse_a, reuse_b)
        c = __builtin_amdgcn_wmma_f32_16x16x4_f32(false, a, false, b,
                                                  (short)0, c, false, false);
    }

    const int rowbase = m0 + ((lane >> 4) << 3);
    float* __restrict__ outcol = H + (size_t)rowbase * DFEAT + n0 + l16;
#pragma unroll
    for (int r = 0; r < 8; ++r) {
        outcol[(size_t)r * DFEAT] = c[r];
    }
}

// ----------------------------------------------------------- edge scatter
// One wave per edge: lane l handles features [4l, 4l+4). 128 feats = 32 lanes x 4.
__global__ void scatter_kernel(const int* __restrict__ row, const int* __restrict__ col,
                               const float* __restrict__ dinv, const float* __restrict__ H,
                               float* __restrict__ agg, int E) {
    const int lane = threadIdx.x & 31;
    const int e = (blockIdx.x * blockDim.x + threadIdx.x) >> 5;
    if (e >= E) return;
    const int r = row[e];
    const int c = col[e];
    const float nrm = dinv[r] * dinv[c];
    const float4 v = reinterpret_cast<const float4*>(H + (size_t)r * DFEAT)[lane];
    float* __restrict__ dst = agg + (size_t)c * DFEAT + lane * 4;
    atomicAdd(dst + 0, nrm * v.x);
    atomicAdd(dst + 1, nrm * v.y);
    atomicAdd(dst + 2, nrm * v.z);
    atomicAdd(dst + 3, nrm * v.w);
}

// ----------------------------------------------------------- BN statistics
// blockDim.x == 128: thread f accumulates sum/sumsq of feature f over a
// node-stripe, then one atomic per block per feature into stats[0:128]=sum,
// stats[128:256]=sumsq.
__global__ void bn_stats(const float* __restrict__ agg, float* __restrict__ stats, int N) {
    const int f = threadIdx.x;
    float s = 0.0f, s2 = 0.0f;
    for (int n = blockIdx.x; n < N; n += gridDim.x) {
        float v = agg[(size_t)n * DFEAT + f];
        s += v;
        s2 += v * v;
    }
    atomicAdd(&stats[f], s);
    atomicAdd(&stats[DFEAT + f], s2);
}

// --------------------------------------------- BN apply + ReLU + residual
__global__ void bn_apply(const float* __restrict__ xin, const float* __restrict__ agg,
                         const float* __restrict__ stats, const float* __restrict__ g,
                         const float* __restrict__ be, float* __restrict__ xout,
                         int total4) {
    const int i = blockIdx.x * blockDim.x + threadIdx.x;
    if (i >= total4) return;
    const float invN = 1.0f / (float)N_NODES;
    const int f0 = (i & 31) << 2;                  // 32 float4 per 128-feature row
    const float4 a  = reinterpret_cast<const float4*>(agg)[i];
    const float4 xv = reinterpret_cast<const float4*>(xin)[i];
    float av[4]  = {a.x, a.y, a.z, a.w};
    float xvv[4] = {xv.x, xv.y, xv.z, xv.w};
    float r[4];
#pragma unroll
    for (int j = 0; j < 4; ++j) {
        const int f = f0 + j;
        const float mu  = stats[f] * invN;
        const float var = stats[DFEAT + f] * invN - mu * mu;   // biased var (jnp.var)
        const float sc  = g[f] * rsqrtf(var + BN_EPS);
        const float hv  = (av[j] - mu) * sc + be[f];
        r[j] = xvv[j] + fmaxf(hv, 0.0f);
    }
    float4 o = {r[0], r[1], r[2], r[3]};
    reinterpret_cast<float4*>(xout)[i] = o;
}

// ------------------------------------------------------------------ launch
extern "C" void kernel_launch(void* const* d_in, const int* in_sizes, int n_in,
                              void* d_out, int out_size, void* d_ws, size_t ws_size,
                              hipStream_t stream) {
    (void)in_sizes; (void)n_in; (void)out_size; (void)ws_size;

    const float* x     = (const float*)d_in[0];   // [N, 128]
    const int*   ei    = (const int*)d_in[1];     // [2, E] (int32 under default JAX x64-off)
    const float* W     = (const float*)d_in[2];   // [3, 128, 128]
    const float* bias  = (const float*)d_in[3];   // [3, 128]
    const float* gamma = (const float*)d_in[4];   // [3, 128]
    const float* beta  = (const float*)d_in[5];   // [3, 128]
    float* out = (float*)d_out;                   // [N, 128]

    const size_t ND = (size_t)N_NODES * DFEAT;
    float* ws    = (float*)d_ws;
    float* x_cur = ws;                // N*D
    float* h     = ws + ND;           // N*D
    float* agg   = ws + 2 * ND;       // N*D
    float* dinv  = ws + 3 * ND;       // N   (holds deg first, then 1/sqrt(deg))
    float* stats = dinv + N_NODES;    // 256 (sum | sumsq)

    const int* row = ei;
    const int* col = ei + N_EDGES;

    hipMemcpyAsync(x_cur, x, ND * sizeof(float), hipMemcpyDeviceToDevice, stream);

    // degree -> dinv
    zero_f32<<<256, 256, 0, stream>>>(dinv, N_NODES);
    degree_kernel<<<(N_EDGES + 255) / 256, 256, 0, stream>>>(row, dinv, N_EDGES);
    dinv_kernel<<<(N_NODES + 255) / 256, 256, 0, stream>>>(dinv, N_NODES);

    const int tiles      = (N_NODES / 16) * (DFEAT / 16);  // 25000 waves
    const int gemmBlocks = tiles / 8;                      // 8 waves / 256-thread block
    const int scatThreads = N_EDGES * 32;                  // one wave per edge

    for (int l = 0; l < N_LAYERS; ++l) {
        gemm_wmma<<<gemmBlocks, 256, 0, stream>>>(x_cur, W + (size_t)l * DFEAT * DFEAT,
                                                  bias + l * DFEAT, h, N_NODES);
        zero_f32<<<2048, 256, 0, stream>>>(agg, (int)ND);
        zero_f32<<<1, 256, 0, stream>>>(stats, 2 * DFEAT);
        scatter_kernel<<<(scatThreads + 255) / 256, 256, 0, stream>>>(row, col, dinv, h,
                                                                      agg, N_EDGES);
        bn_stats<<<512, DFEAT, 0, stream>>>(agg, stats, N_NODES);
        float* xo = (l == N_LAYERS - 1) ? out : x_cur;
        bn_apply<<<((int)(ND / 4) + 255) / 256, 256, 0, stream>>>(
            x_cur, agg, stats, gamma + l * DFEAT, beta + l * DFEAT, xo, (int)(ND / 4));
    }
}